// Self_Attention_11244224380956
// MI455X (gfx1250) — compile-verified
//
#include <hip/hip_runtime.h>
#include <math.h>

// Problem constants from the reference
#define BB 32
#define TT 4096
#define DD 1024
#define ATT_SCALE (1.0f / 32.0f)   // 1/sqrt(1024)

typedef __attribute__((ext_vector_type(2))) float v2f;
typedef __attribute__((ext_vector_type(8))) float v8f;

// ---------------------------------------------------------------------------
// Kernel 1: scores[b,t] = dot(q[b,:], K[b,t,:])  via V_WMMA_F32_16X16X4_F32
// One wave handles a 16-row tile of K; block = 8 waves = 128 rows.
// A (16x4 f32): lane m (0..15) vgpr0/1 = cols kc+0/kc+1 ; lane 16+m = kc+2/kc+3
//   -> per-lane float2 load at column kc + 2*half.
// B (4x16 f32): q chunk replicated across all 16 columns. Row striping rule
//   (vgpr j, half h -> row j + 2h) means the same float2 offset works.
// D: all 16 columns equal the dot product; vgpr j at half h = row j + 8h.
// ---------------------------------------------------------------------------
__global__ __launch_bounds__(256) void scores_wmma_kernel(
    const float* __restrict__ q,      // [B, D]
    const float* __restrict__ K,      // [B, T, D]
    float* __restrict__ scores)       // ws: [B, T] (unscaled dot products)
{
    __shared__ float q_lds[DD];

    const int tilesPerBatch = TT / 128;                 // 32
    const int b     = blockIdx.x / tilesPerBatch;
    const int t_blk = (blockIdx.x % tilesPerBatch) * 128;
    const int tid   = threadIdx.x;

    // Stage q[b,:] into LDS, coalesced float4 (256 threads x 4 floats)
    {
        const float4* qg = (const float4*)(q + (size_t)b * DD);
        ((float4*)q_lds)[tid] = qg[tid];
    }
    __syncthreads();

    const int wave = tid >> 5;        // 0..7
    const int lane = tid & 31;
    const int half = lane >> 4;       // 0 or 1
    const int m    = lane & 15;       // row within the 16-row tile
    const int t0   = t_blk + wave * 16;

    const float* krow = K + ((size_t)b * TT + (size_t)(t0 + m)) * DD + 2 * half;
    const float* qrow = q_lds + 2 * half;

    v8f acc = {};
#pragma unroll 8
    for (int kc = 0; kc < DD; kc += 4) {
        v2f a  = *(const v2f*)(krow + kc);   // K tile operand (A matrix)
        v2f bq = *(const v2f*)(qrow + kc);   // q chunk broadcast (B matrix)
        acc = __builtin_amdgcn_wmma_f32_16x16x4_f32(
            /*neg_a=*/false, a, /*neg_b=*/false, bq,
            /*c_mod=*/(short)0, acc, /*reuse_a=*/false, /*reuse_b=*/false);
    }

    // Every column of D is identical; lane 0 holds rows 0..7, lane 16 rows 8..15.
    if (m == 0) {
        float* srow = scores + (size_t)b * TT + t0 + 8 * half;
#pragma unroll
        for (int j = 0; j < 8; ++j) srow[j] = acc[j];
    }
}

// ---------------------------------------------------------------------------
// Kernel 2: per-batch softmax over T=4096 scores.  One block per batch.
// ---------------------------------------------------------------------------
__global__ __launch_bounds__(256) void softmax_kernel(
    const float* __restrict__ scores,   // ws: [B, T]
    float* __restrict__ attn)           // out region: [B, T]
{
    __shared__ float red[256];
    const int b   = blockIdx.x;
    const int tid = threadIdx.x;
    const float* srow = scores + (size_t)b * TT;

    float s[16];
    float lmax = -INFINITY;
#pragma unroll
    for (int i = 0; i < 16; ++i) {
        s[i] = srow[tid + 256 * i] * ATT_SCALE;
        lmax = fmaxf(lmax, s[i]);
    }
    red[tid] = lmax;
    __syncthreads();
    for (int off = 128; off > 0; off >>= 1) {
        if (tid < off) red[tid] = fmaxf(red[tid], red[tid + off]);
        __syncthreads();
    }
    const float gmax = red[0];
    __syncthreads();

    float lsum = 0.f;
#pragma unroll
    for (int i = 0; i < 16; ++i) {
        s[i] = __expf(s[i] - gmax);
        lsum += s[i];
    }
    red[tid] = lsum;
    __syncthreads();
    for (int off = 128; off > 0; off >>= 1) {
        if (tid < off) red[tid] += red[tid + off];
        __syncthreads();
    }
    const float inv = 1.0f / red[0];

    float* arow = attn + (size_t)b * TT;
#pragma unroll
    for (int i = 0; i < 16; ++i) arow[tid + 256 * i] = s[i] * inv;
}

// ---------------------------------------------------------------------------
// Kernel 3: partial[b,ts,:] = sum_{t in slice ts} attn[b,t] * V[b,t,:]
// Block = (b, ts); per t the whole block streams V[b,t,:] (4KB, float4,
// fully coalesced). 512 blocks -> enough waves to saturate HBM.
// ---------------------------------------------------------------------------
#define TSPLIT 16
__global__ __launch_bounds__(256) void av_partial_kernel(
    const float* __restrict__ attn,     // [B, T]
    const float* __restrict__ V,        // [B, T, D]
    float* __restrict__ partial)        // ws: [B, TSPLIT, D]
{
    const int b      = blockIdx.x / TSPLIT;
    const int ts     = blockIdx.x % TSPLIT;
    const int tid    = threadIdx.x;
    const int tchunk = TT / TSPLIT;     // 256

    const float*  arow  = attn + (size_t)b * TT + (size_t)ts * tchunk;
    const float4* vbase = (const float4*)(V + ((size_t)b * TT + (size_t)ts * tchunk) * DD) + tid;

    float4 acc = {0.f, 0.f, 0.f, 0.f};
    for (int t = 0; t < tchunk; ++t) {
        const float w = arow[t];
        float4 v = vbase[(size_t)t * (DD / 4)];
        acc.x += w * v.x;
        acc.y += w * v.y;
        acc.z += w * v.z;
        acc.w += w * v.w;
    }
    ((float4*)(partial + ((size_t)(b * TSPLIT + ts)) * DD))[tid] = acc;
}

// ---------------------------------------------------------------------------
// Kernel 4: out[b,d] = sum_ts partial[b,ts,d]   (deterministic reduction)
// ---------------------------------------------------------------------------
__global__ __launch_bounds__(256) void reduce_out_kernel(
    const float* __restrict__ partial,  // [B, TSPLIT, D]
    float* __restrict__ out)            // [B, D]
{
    const int idx = blockIdx.x * 256 + threadIdx.x;   // 0 .. B*D-1
    const int b = idx / DD;
    const int d = idx % DD;
    float s = 0.f;
#pragma unroll
    for (int ts = 0; ts < TSPLIT; ++ts)
        s += partial[((size_t)(b * TSPLIT + ts)) * DD + d];
    out[idx] = s;
}

// ---------------------------------------------------------------------------
extern "C" void kernel_launch(void* const* d_in, const int* in_sizes, int n_in,
                              void* d_out, int out_size, void* d_ws, size_t ws_size,
                              hipStream_t stream) {
    const float* q = (const float*)d_in[0];   // [B, D]
    const float* K = (const float*)d_in[1];   // [B, T, D]
    const float* V = (const float*)d_in[2];   // [B, T, D]

    float* out  = (float*)d_out;                       // [B, D]   (32768 f32)
    float* attn = (float*)d_out + (size_t)BB * DD;     // [B, T]   (131072 f32)

    float* scores  = (float*)d_ws;                     // B*T floats (512 KB)
    float* partial = (float*)d_ws + (size_t)BB * TT;   // B*TSPLIT*D floats (2 MB)

    scores_wmma_kernel<<<BB * (TT / 128), 256, 0, stream>>>(q, K, scores);
    softmax_kernel<<<BB, 256, 0, stream>>>(scores, attn);
    av_partial_kernel<<<BB * TSPLIT, 256, 0, stream>>>(attn, V, partial);
    reduce_out_kernel<<<(BB * DD) / 256, 256, 0, stream>>>(partial, out);
}